// SimpleNetwork_36541581754801
// MI455X (gfx1250) — compile-verified
//
#include <hip/hip_runtime.h>

typedef float v2f __attribute__((ext_vector_type(2)));
typedef float v8f __attribute__((ext_vector_type(8)));
typedef int   v4i __attribute__((ext_vector_type(4)));

// ---------------------------------------------------------------------------
// Kernel 1: zero the aggregation workspace (float4-vectorized).
// ---------------------------------------------------------------------------
__global__ void zero_ws_kernel(float* __restrict__ ws, int n_floats) {
    int i = (blockIdx.x * blockDim.x + threadIdx.x) * 4;
    if (i + 3 < n_floats) {
        float4 z = make_float4(0.f, 0.f, 0.f, 0.f);
        *reinterpret_cast<float4*>(ws + i) = z;
    } else {
        for (int k = i; k < n_floats; ++k) ws[k] = 0.f;
    }
}

// ---------------------------------------------------------------------------
// Kernel 2: edge scatter-add.  aggr[dst[e]] += x[src[e]]  (D = 2).
// The 128 MB src/dst index stream is read exactly once -> non-temporal
// 128-bit loads (don't pollute WGP$).  x (4 MB) and aggr (4 MB) are
// L2-resident and reused across all 16M edges -> default RT policy for the
// gathers, hardware global_atomic_add_f32 for the scatter.
// ---------------------------------------------------------------------------
__global__ void edge_scatter_kernel(const int* __restrict__ src,
                                    const int* __restrict__ dst,
                                    const float* __restrict__ x,
                                    float* __restrict__ aggr,
                                    int n_edges) {
    int t  = blockIdx.x * blockDim.x + threadIdx.x;
    int e0 = t * 4;
    if (e0 + 3 < n_edges) {
        v4i s = __builtin_nontemporal_load(reinterpret_cast<const v4i*>(src + e0));
        v4i d = __builtin_nontemporal_load(reinterpret_cast<const v4i*>(dst + e0));

        float2 xs0 = *reinterpret_cast<const float2*>(x + (size_t)s.x * 2);
        float2 xs1 = *reinterpret_cast<const float2*>(x + (size_t)s.y * 2);
        float2 xs2 = *reinterpret_cast<const float2*>(x + (size_t)s.z * 2);
        float2 xs3 = *reinterpret_cast<const float2*>(x + (size_t)s.w * 2);

        unsafeAtomicAdd(&aggr[(size_t)d.x * 2 + 0], xs0.x);
        unsafeAtomicAdd(&aggr[(size_t)d.x * 2 + 1], xs0.y);
        unsafeAtomicAdd(&aggr[(size_t)d.y * 2 + 0], xs1.x);
        unsafeAtomicAdd(&aggr[(size_t)d.y * 2 + 1], xs1.y);
        unsafeAtomicAdd(&aggr[(size_t)d.z * 2 + 0], xs2.x);
        unsafeAtomicAdd(&aggr[(size_t)d.z * 2 + 1], xs2.y);
        unsafeAtomicAdd(&aggr[(size_t)d.w * 2 + 0], xs3.x);
        unsafeAtomicAdd(&aggr[(size_t)d.w * 2 + 1], xs3.y);
    } else {
        for (int e = e0; e < n_edges; ++e) {
            int s = src[e], d = dst[e];
            float2 xs = *reinterpret_cast<const float2*>(x + (size_t)s * 2);
            unsafeAtomicAdd(&aggr[(size_t)d * 2 + 0], xs.x);
            unsafeAtomicAdd(&aggr[(size_t)d * 2 + 1], xs.y);
        }
    }
}

// ---------------------------------------------------------------------------
// Kernel 3: fused epilogue via V_WMMA_F32_16X16X4_F32.
//   out = [aggr | x] (16x4) @ [W_rel^T ; W_root^T] (4x16, cols 0..1 live) + b
// A layout (32-bit A 16x4): lanes 0-15 hold K=0,1 (aggr), lanes 16-31 hold
// K=2,3 (x); M = lane % 16.  B layout mirrors: VGPR0/1 = rows {0,1} for
// lanes 0-15 and rows {2,3} for lanes 16-31, column N = lane % 16.
// C/D layout: VGPR r, lanes 0-15 -> (M=r, N=lane); lanes 16-31 -> (M=r+8).
// All tile-size decisions are wave-uniform so EXEC is all-ones at the WMMA;
// full tiles take a branch-free load/store path.
// ---------------------------------------------------------------------------
__global__ void node_epilogue_wmma(const float* __restrict__ x,
                                   const float* __restrict__ aggr,
                                   const float* __restrict__ W_rel,
                                   const float* __restrict__ b_rel,
                                   const float* __restrict__ W_root,
                                   float* __restrict__ out,
                                   int n_nodes) {
    const int wave      = threadIdx.x >> 5;
    const int lane      = threadIdx.x & 31;
    const int num_tiles = (n_nodes + 15) >> 4;
    const int tile      = blockIdx.x * (blockDim.x >> 5) + wave;
    if (tile >= num_tiles) return;             // wave-uniform

    const bool full = (tile * 16 + 16) <= n_nodes;  // wave-uniform
    const int  n    = lane & 15;   // column (B,C,D) / row M (A)
    const int  half = lane >> 4;   // 0: K={0,1}, 1: K={2,3}

    // ---- A: per-lane pair of K values for row M = tile*16 + n ----
    const int row = tile * 16 + n;
    const float* __restrict__ srcA = half ? x : aggr;
    v2f a;
    a.x = 0.f; a.y = 0.f;
    if (full) {                                 // uniform: no EXEC churn
        float2 v = *reinterpret_cast<const float2*>(srcA + (size_t)row * 2);
        a.x = v.x; a.y = v.y;
    } else if (row < n_nodes) {
        float2 v = *reinterpret_cast<const float2*>(srcA + (size_t)row * 2);
        a.x = v.x; a.y = v.y;
    }

    // ---- B: B[k][n] = W[n][k mod 2], only columns n < 2 are non-zero ----
    v2f b;
    b.x = 0.f; b.y = 0.f;
    if (n < 2) {
        const float* __restrict__ W = half ? W_root : W_rel;  // row-major 2x2
        b.x = W[n * 2 + 0];
        b.y = W[n * 2 + 1];
    }

    // ---- C: broadcast bias into columns 0..1 ----
    const float cv = (n < 2) ? b_rel[n] : 0.f;
    v8f c;
#pragma unroll
    for (int r = 0; r < 8; ++r) c[r] = cv;

    // ---- D = A x B + C  (v_wmma_f32_16x16x4_f32) ----
    v8f d = __builtin_amdgcn_wmma_f32_16x16x4_f32(
        /*neg_a=*/false, a, /*neg_b=*/false, b,
        /*c_mod=*/(short)0, c, /*reuse_a=*/false, /*reuse_b=*/false);

    // ---- store: lane n<2 in each half holds out[tile*16 + half*8 + r][n] ----
    const int m0 = tile * 16 + half * 8;
    if (full) {                                 // uniform fast path
        if (n < 2) {
            float* __restrict__ p = out + (size_t)m0 * 2 + n;
#pragma unroll
            for (int r = 0; r < 8; ++r) p[(size_t)r * 2] = d[r];
        }
    } else {
        if (n < 2) {
#pragma unroll
            for (int r = 0; r < 8; ++r) {
                int m = m0 + r;
                if (m < n_nodes) out[(size_t)m * 2 + n] = d[r];
            }
        }
    }
}

// ---------------------------------------------------------------------------
// Launch
// ---------------------------------------------------------------------------
extern "C" void kernel_launch(void* const* d_in, const int* in_sizes, int n_in,
                              void* d_out, int out_size, void* d_ws, size_t ws_size,
                              hipStream_t stream) {
    const float* x          = (const float*)d_in[0];  // [N, 2]
    const int*   edge_index = (const int*)d_in[1];    // [2, E]
    const float* W_rel      = (const float*)d_in[2];  // [2, 2]
    const float* b_rel      = (const float*)d_in[3];  // [2]
    const float* W_root     = (const float*)d_in[4];  // [2, 2]
    float*       out        = (float*)d_out;          // [N, 2]
    float*       aggr       = (float*)d_ws;           // [N, 2] scratch

    const int n_nodes = in_sizes[0] / 2;
    const int n_edges = in_sizes[1] / 2;
    const int* src = edge_index;
    const int* dst = edge_index + n_edges;

    // 1) zero aggregation buffer (deterministic per call)
    {
        int n_floats = n_nodes * 2;
        int threads  = (n_floats + 3) / 4;
        int blocks   = (threads + 255) / 256;
        zero_ws_kernel<<<blocks, 256, 0, stream>>>(aggr, n_floats);
    }

    // 2) edge scatter-add (4 edges per thread, NT index stream)
    {
        int threads = (n_edges + 3) / 4;
        int blocks  = (threads + 255) / 256;
        edge_scatter_kernel<<<blocks, 256, 0, stream>>>(src, dst, x, aggr, n_edges);
    }

    // 3) WMMA epilogue: 16 rows per wave, 8 waves per 256-thread block
    {
        int num_tiles = (n_nodes + 15) / 16;
        int blocks    = (num_tiles + 7) / 8;
        node_epilogue_wmma<<<blocks, 256, 0, stream>>>(x, aggr, W_rel, b_rel,
                                                       W_root, out, n_nodes);
    }
}